// FreqHash_2671469658638
// MI455X (gfx1250) — compile-verified
//
#include <hip/hip_runtime.h>
#include <stdint.h>

// Problem constants (from reference)
#define NFREQ   6
#define NB      36        // NFREQ*2*3 encodings
#define NCH     48
#define RES     256
#define NPTS    32768
#define PCHUNK  512       // points per block
#define ROWP    257       // padded LDS row stride in floats (TDM pad: +1 DWORD / 256 DWORDs)
#define BLOCK   256

typedef unsigned int u32x4 __attribute__((ext_vector_type(4)));
typedef int          i32x8 __attribute__((ext_vector_type(8)));
typedef int          i32x4 __attribute__((ext_vector_type(4)));

__global__ __launch_bounds__(BLOCK)
void freqhash_kernel(const float* __restrict__ points,
                     const float* __restrict__ cv,
                     const int*   __restrict__ scale_p,
                     float*       __restrict__ out)
{
    // 48 channels x 257 floats (padded rows) = 49,344 B + 6 KB staging < 64 KB
    __shared__ float sTbl[NCH * ROWP];
    __shared__ float sEnc[PCHUNK];
    __shared__ float sW1[PCHUNK];
    __shared__ int   sI01[PCHUNK];

    const int b     = blockIdx.y;   // which encoding / table slice
    const int chunk = blockIdx.x;   // which block of 512 points
    const int tid   = threadIdx.x;

    // ---------------------------------------------------------------
    // Issue TDM DMA: cv[b][0:48][0:256] (contiguous 48 KB) -> LDS,
    // with 1-DWORD padding after every 256 DWORDs so channel rows land
    // at stride 257 floats (bank-conflict-free gathers later).
    // Wave 0 only issues; EXEC is ignored by TDM but branch is wave-uniform.
    // ---------------------------------------------------------------
    if (tid < 32) {
        unsigned long long ga = (unsigned long long)(uintptr_t)(cv + (size_t)b * (NCH * RES));
        unsigned lds_off = (unsigned)(uintptr_t)&sTbl[0]; // low 32 bits = LDS offset

        // D# group 0: count=1 | lds_addr | global_addr[56:0] | type=2
        u32x4 g0 = { 1u,
                     lds_off,
                     (unsigned)(ga & 0xFFFFFFFFu),
                     (unsigned)((ga >> 32) & 0x01FFFFFFu) | (2u << 30) };

        // D# group 1:
        //  w0: wg_mask=0 | data_size=2 (4B) <<16 | pad_enable<<20 | pad_interval=7 (256 DW)<<22 | pad_amount=0 (1 DW)
        //  tensor_dim0 = tile_dim0 = tensor_dim0_stride = 12288 elements; tensor_dim1 = tile_dim1 = 1
        i32x8 g1 = { (int)((2u << 16) | (1u << 20) | (7u << 22)),   // 0x01D20000
                     (int)0x30000000,                               // tensor_dim0[15:0]=0x3000 in bits 63:48
                     (int)0x00010000,                               // tensor_dim0 hi=0 | tensor_dim1[15:0]=1
                     (int)0x30000000,                               // tensor_dim1 hi=0 | tile_dim0=0x3000
                     1,                                             // tile_dim1=1 | tile_dim2=0
                     (int)(NCH * RES),                              // tensor_dim0_stride[31:0]=12288
                     0, 0 };                                        // stride hi / dim1_stride = 0

        i32x4 g2 = { 0, 0, 0, 0 };           // <=2D tensor: groups 2/3 unused
        i32x4 g3 = { 0, 0, 0, 0 };
        i32x8 g4 = { 0, 0, 0, 0, 0, 0, 0, 0 };  // extra group (clang-23 6-arg form), unused

        __builtin_amdgcn_tensor_load_to_lds(g0, g1, g2, g3, g4, 0);
    }

    // ---------------------------------------------------------------
    // Phase 1 (overlaps DMA): per-point encoding + interp coefficients.
    //   b = f*6 + s*3 + d ; freqs = 1,2,4,8,16,32 ; y=(enc+1)/2*(RES-1)
    // ---------------------------------------------------------------
    const int   d   = b % 3;
    const int   sc  = (b / 3) % 2;          // 0 = sin, 1 = cos
    const int   f   = b / 6;
    const float mul = (float)(1 << f) / (float)(*scale_p);

    for (int i = tid; i < PCHUNK; i += BLOCK) {
        const int   n = chunk * PCHUNK + i;
        const float x = points[n * 3 + d] * mul;
        const float e = sc ? __cosf(x) : __sinf(x);
        const float y = (e + 1.0f) * (0.5f * (float)(RES - 1));
        const float y0 = floorf(y);
        int i0 = (int)y0;
        i0 = i0 < 0 ? 0 : (i0 > RES - 1 ? RES - 1 : i0);
        const int i1 = (i0 + 1 > RES - 1) ? RES - 1 : i0 + 1;
        sEnc[i] = e;
        sW1[i]  = y - y0;
        sI01[i] = i0 | (i1 << 16);
    }

    if (tid < 32) {
        __builtin_amdgcn_s_wait_tensorcnt(0);   // DMA complete for issuing wave
    }
    __syncthreads();                            // LDS visible to whole workgroup

    // ---------------------------------------------------------------
    // Phase 2: one output per (n, c): 2 conflict-free LDS gathers + fma.
    //   out[n, c*36 + b]
    // ---------------------------------------------------------------
    for (int p = tid; p < PCHUNK * NCH; p += BLOCK) {
        const int n  = p / NCH;
        const int c  = p - n * NCH;
        const int ii = sI01[n];
        const float w1 = sW1[n];
        const float e  = sEnc[n];
        const float v0 = sTbl[c * ROWP + (ii & 0xFFFF)];
        const float v1 = sTbl[c * ROWP + (ii >> 16)];
        const float r  = fmaf(v1 - v0, w1, v0) + e;
        const size_t gn = (size_t)(chunk * PCHUNK + n);
        out[gn * (size_t)(NCH * NB) + (size_t)(c * NB + b)] = r;
    }
}

extern "C" void kernel_launch(void* const* d_in, const int* in_sizes, int n_in,
                              void* d_out, int out_size, void* d_ws, size_t ws_size,
                              hipStream_t stream)
{
    const float* points = (const float*)d_in[0];   // [32768,3] f32
    const float* cv     = (const float*)d_in[1];   // [36,48,256,1] f32
    const int*   scale  = (const int*)d_in[2];     // scalar int (3)
    float*       out    = (float*)d_out;           // [32768, 1728] f32

    dim3 grid(NPTS / PCHUNK, NB);
    freqhash_kernel<<<grid, BLOCK, 0, stream>>>(points, cv, scale, out);
}